// GCLayer_21672404975674
// MI455X (gfx1250) — compile-verified
//
#include <hip/hip_runtime.h>
#include <math.h>

#define Nn 512

typedef __attribute__((ext_vector_type(2))) float v2f;
typedef __attribute__((ext_vector_type(8))) float v8f;

// One wave computes a 16x64 fp32 tile: C += A(16x64) @ W(64x64), using
// V_WMMA_F32_16X16X4_F32 (fp32 A/B, fp32 accum -> exact vs reference).
// A layout per ISA: lane<16 -> row=lane, k-offsets {0,1}; lane>=16 -> {2,3}.
// B layout mirrors A's K striping; C/D: row=(lane>>4)*8+v, col=(lane&15)+16t.
__device__ __forceinline__ void gemm16(const float* __restrict__ A, int lda,
                                       const float* __restrict__ W,
                                       v8f acc[4]) {
  const int lane = threadIdx.x & 31;
  const int r  = lane & 15;
  const int kh = (lane >> 4) * 2;
  const int c  = lane & 15;
#pragma unroll
  for (int kk = 0; kk < 64; kk += 4) {
    v2f a;
    a.x = A[r * lda + kk + kh];
    a.y = A[r * lda + kk + kh + 1];
#pragma unroll
    for (int t = 0; t < 4; ++t) {
      v2f b;
      b.x = W[(kk + kh) * 64 + t * 16 + c];
      b.y = W[(kk + kh + 1) * 64 + t * 16 + c];
      acc[t] = __builtin_amdgcn_wmma_f32_16x16x4_f32(
          false, a, false, b, (short)0, acc[t], false, false);
    }
  }
}

// Stage 1: h = x@Wl + bl ; hr_m = h@Wm1[:64] ; hl_a = h@Wa1[:64] ; hr_a = h@Wa1[64:]
// One wave per 16-row tile, 4 waves/block. LDS round-trip converts h from the
// WMMA C layout back to row-major so it can feed the next GEMMs as A.
__global__ __launch_bounds__(128) void gc_k1(
    const float* __restrict__ x, const float* __restrict__ Wl,
    const float* __restrict__ bl, const float* __restrict__ Wm1,
    const float* __restrict__ Wa1, float* __restrict__ h,
    float* __restrict__ hrm, float* __restrict__ hla, float* __restrict__ hra) {
  __shared__ float lds[4 * 16 * 64];
  const int wave = threadIdx.x >> 5;
  const int lane = threadIdx.x & 31;
  const int row0 = (blockIdx.x * 4 + wave) * 16;
  const int rb = (lane >> 4) * 8;
  const int c = lane & 15;
  float* myl = &lds[wave * 1024];

  v8f acc[4] = {};
  gemm16(x + (size_t)row0 * 64, 64, Wl, acc);
#pragma unroll
  for (int t = 0; t < 4; ++t) {
    float bv = bl[t * 16 + c];
#pragma unroll
    for (int v = 0; v < 8; ++v) {
      float val = acc[t][v] + bv;
      h[((size_t)row0 + rb + v) * 64 + t * 16 + c] = val;
      myl[(rb + v) * 64 + t * 16 + c] = val;
    }
  }
  __syncthreads();
  v8f a1[4] = {}, a2[4] = {}, a3[4] = {};
  gemm16(myl, 64, Wm1, a1);
  gemm16(myl, 64, Wa1, a2);
  gemm16(myl, 64, Wa1 + 64 * 64, a3);
#pragma unroll
  for (int t = 0; t < 4; ++t)
#pragma unroll
    for (int v = 0; v < 8; ++v) {
      size_t idx = ((size_t)row0 + rb + v) * 64 + t * 16 + c;
      hrm[idx] = a1[t][v];
      hla[idx] = a2[t][v];
      hra[idx] = a3[t][v];
    }
}

// Stage 2: per (b,i): acc[c] = sum_j att_ij * relu(hr_m[j,c] + adj_ij*w[c] + bm1[c])
// att_ij = sigmoid(sum_c silu(hl_a[i,c]+hr_a[j,c]+ba1[c]) * Wa2[c] + ba2) * adj_ij
// One block per (b,i); 8 waves x 64 neighbors each; lane owns channels c, c+32.
// All per-j vectors are L2-resident (whole ws ~2.6MB << 192MB L2).
__global__ __launch_bounds__(256) void gc_k2(
    const float* __restrict__ adj, const float* __restrict__ hrm,
    const float* __restrict__ hla, const float* __restrict__ hra,
    const float* __restrict__ wlast, const float* __restrict__ bm1,
    const float* __restrict__ ba1, const float* __restrict__ Wa2,
    const float* __restrict__ ba2p, float* __restrict__ accOut,
    float* __restrict__ attOut) {
  const int bi = blockIdx.x;        // b*N + i
  const int b = bi >> 9;
  const int tid = threadIdx.x;
  const int wave = tid >> 5;
  const int lane = tid & 31;
  const int c0 = lane, c1 = lane + 32;
  const float w0 = wlast[c0], w1 = wlast[c1];
  const float bm0 = bm1[c0], bm1v = bm1[c1];
  const float ba0 = ba1[c0], ba1v = ba1[c1];
  const float wa0 = Wa2[c0], wa1 = Wa2[c1];
  const float ba2 = ba2p[0];
  const float hl0 = hla[(size_t)bi * 64 + c0];
  const float hl1 = hla[(size_t)bi * 64 + c1];
  const float* adjrow = adj + (size_t)bi * Nn;
  const float* hrmb = hrm + (size_t)b * Nn * 64;
  const float* hrab = hra + (size_t)b * Nn * 64;

  float acc0 = 0.f, acc1 = 0.f, asum = 0.f;
  for (int j = wave * 64; j < wave * 64 + 64; ++j) {
    float av = adjrow[j];
    float m0 = fmaxf(hrmb[(size_t)j * 64 + c0] + av * w0 + bm0, 0.f);
    float m1 = fmaxf(hrmb[(size_t)j * 64 + c1] + av * w1 + bm1v, 0.f);
    float z0 = hl0 + hrab[(size_t)j * 64 + c0] + ba0;
    float z1 = hl1 + hrab[(size_t)j * 64 + c1] + ba1v;
    float s0 = z0 / (1.f + __expf(-z0));   // silu
    float s1 = z1 / (1.f + __expf(-z1));
    float d = s0 * wa0 + s1 * wa1;
#pragma unroll
    for (int m = 16; m; m >>= 1) d += __shfl_xor(d, m, 32);  // wave32 dot
    float att = av / (1.f + __expf(-(d + ba2)));             // sigmoid * adj
    acc0 += att * m0;
    acc1 += att * m1;
    asum += att;
  }
  __shared__ float red[8 * 64];
  __shared__ float reda[8];
  red[wave * 64 + c0] = acc0;
  red[wave * 64 + c1] = acc1;
  if (lane == 0) reda[wave] = asum;
  __syncthreads();
  if (tid < 64) {
    float s = 0.f;
#pragma unroll
    for (int w = 0; w < 8; ++w) s += red[w * 64 + tid];
    accOut[(size_t)bi * 64 + tid] = s;
    if (tid == 0) {
      float sa = 0.f;
#pragma unroll
      for (int w = 0; w < 8; ++w) sa += reda[w];
      attOut[bi] = sa;
    }
  }
}

// Stage 3: msg = acc@Wm2 + bm2*attSum ; t = relu(msg@Wo1+bo1) ; u = t@Wo2+bo2
// h += u ; LayerNorm (row stats via shfl_xor within 16-lane half groups) ; relu
__global__ __launch_bounds__(128) void gc_k3(
    const float* __restrict__ accIn, const float* __restrict__ attSum,
    const float* __restrict__ h, const float* __restrict__ Wm2,
    const float* __restrict__ bm2, const float* __restrict__ Wo1,
    const float* __restrict__ bo1, const float* __restrict__ Wo2,
    const float* __restrict__ bo2, const float* __restrict__ g,
    const float* __restrict__ beta, float* __restrict__ out) {
  __shared__ float lds[4 * 16 * 64];
  const int wave = threadIdx.x >> 5;
  const int lane = threadIdx.x & 31;
  const int row0 = (blockIdx.x * 4 + wave) * 16;
  const int rb = (lane >> 4) * 8;
  const int c = lane & 15;
  float* myl = &lds[wave * 1024];

  v8f m[4] = {};
  gemm16(accIn + (size_t)row0 * 64, 64, Wm2, m);
  float ats[8];
#pragma unroll
  for (int v = 0; v < 8; ++v) ats[v] = attSum[row0 + rb + v];
#pragma unroll
  for (int t = 0; t < 4; ++t) {
    float bv = bm2[t * 16 + c];
#pragma unroll
    for (int v = 0; v < 8; ++v)
      myl[(rb + v) * 64 + t * 16 + c] = m[t][v] + bv * ats[v];
  }
  __syncthreads();
  v8f t1[4] = {};
  gemm16(myl, 64, Wo1, t1);
  __syncthreads();
#pragma unroll
  for (int t = 0; t < 4; ++t) {
    float bv = bo1[t * 16 + c];
#pragma unroll
    for (int v = 0; v < 8; ++v)
      myl[(rb + v) * 64 + t * 16 + c] = fmaxf(t1[t][v] + bv, 0.f);
  }
  __syncthreads();
  v8f t2[4] = {};
  gemm16(myl, 64, Wo2, t2);

  float hf[4][8];
#pragma unroll
  for (int t = 0; t < 4; ++t) {
    float bv = bo2[t * 16 + c];
#pragma unroll
    for (int v = 0; v < 8; ++v)
      hf[t][v] = t2[t][v] + bv + h[((size_t)row0 + rb + v) * 64 + t * 16 + c];
  }
#pragma unroll
  for (int v = 0; v < 8; ++v) {
    float s = hf[0][v] + hf[1][v] + hf[2][v] + hf[3][v];
    float q = hf[0][v] * hf[0][v] + hf[1][v] * hf[1][v] +
              hf[2][v] * hf[2][v] + hf[3][v] * hf[3][v];
#pragma unroll
    for (int msk = 8; msk; msk >>= 1) {   // reduce over 16 lanes of half-wave
      s += __shfl_xor(s, msk, 32);
      q += __shfl_xor(q, msk, 32);
    }
    float mu = s * (1.f / 64.f);
    float var = q * (1.f / 64.f) - mu * mu;
    float rstd = rsqrtf(var + 1e-5f);
#pragma unroll
    for (int t = 0; t < 4; ++t) {
      int col = t * 16 + c;
      float o = (hf[t][v] - mu) * rstd * g[col] + beta[col];
      out[((size_t)row0 + rb + v) * 64 + col] = fmaxf(o, 0.f);
    }
  }
}

extern "C" void kernel_launch(void* const* d_in, const int* in_sizes, int n_in,
                              void* d_out, int out_size, void* d_ws, size_t ws_size,
                              hipStream_t stream) {
  const float* x   = (const float*)d_in[0];
  const float* adj = (const float*)d_in[1];
  const float* Wl  = (const float*)d_in[2];
  const float* bl  = (const float*)d_in[3];
  const float* Wm1 = (const float*)d_in[4];
  const float* bm1 = (const float*)d_in[5];
  const float* Wm2 = (const float*)d_in[6];
  const float* bm2 = (const float*)d_in[7];
  const float* Wa1 = (const float*)d_in[8];
  const float* ba1 = (const float*)d_in[9];
  const float* Wa2 = (const float*)d_in[10];
  const float* ba2 = (const float*)d_in[11];
  const float* Wo1 = (const float*)d_in[12];
  const float* bo1 = (const float*)d_in[13];
  const float* Wo2 = (const float*)d_in[14];
  const float* bo2 = (const float*)d_in[15];
  const float* g   = (const float*)d_in[16];
  const float* bet = (const float*)d_in[17];

  float* ws = (float*)d_ws;
  const size_t R = 4 * 512;  // 2048 rows
  float* h    = ws;
  float* hrm  = ws + R * 64;
  float* hla  = ws + 2 * R * 64;
  float* hra  = ws + 3 * R * 64;
  float* accW = ws + 4 * R * 64;
  float* attS = ws + 5 * R * 64;
  float* out  = (float*)d_out;

  gc_k1<<<32, 128, 0, stream>>>(x, Wl, bl, Wm1, Wa1, h, hrm, hla, hra);
  gc_k2<<<2048, 256, 0, stream>>>(adj, hrm, hla, hra, Wm1 + 64 * 64, bm1, ba1,
                                  Wa2, ba2, accW, attS);
  gc_k3<<<32, 128, 0, stream>>>(accW, attS, h, Wm2, bm2, Wo1, bo1, Wo2, bo2,
                                g, bet, out);
  // Tuple output: (out, adj) concatenated flat.
  hipMemcpyAsync(out + R * 64, adj, (size_t)4 * 512 * 512 * sizeof(float),
                 hipMemcpyDeviceToDevice, stream);
}